// RNNTLoss_88227218195237
// MI455X (gfx1250) — compile-verified
//
#include <hip/hip_runtime.h>
#include <hip/hip_bf16.h>
#include <math.h>

#define B_  8
#define T_  256
#define U_  64
#define U1_ 65
#define V1_ 1024
#define ROWS_ (B_ * T_ * U1_)   // 133120 rows of 1024 floats

typedef __attribute__((ext_vector_type(2))) float v2f;
typedef __attribute__((ext_vector_type(8))) float v8f;

__device__ __forceinline__ float logaddexpf_(float a, float b) {
    float m = fmaxf(a, b);
    float d = fabsf(a - b);
    return m + log1pf(expf(-d));
}

// ---------------------------------------------------------------------------
// Kernel 1: streaming log-softmax. 512 threads = 16 waves, one row per wave.
// Each lane holds 32 floats of the row in VGPRs (8x b128 coalesced loads).
// Wave max via shfl_xor; per-lane sum-of-exp partials go to LDS; wave 0
// reduces the 16x32 partial matrix with 8x v_wmma_f32_16x16x4_f32 (B = ones).
// ---------------------------------------------------------------------------
__global__ __launch_bounds__(512) void rnnt_logsoftmax_kernel(
    const float* __restrict__ logits, const int* __restrict__ y,
    float* __restrict__ lp_blank, float* __restrict__ lp_label)
{
    __shared__ float sP[16 * 33];            // padded stride to spread banks
    __shared__ float sMax[16], sBlank[16], sLab[16];
    __shared__ float sD[8 * 32];             // D-matrix bounce buffer

    const int tid  = threadIdx.x;
    const int wave = tid >> 5;
    const int lane = tid & 31;
    const int row0 = blockIdx.x * 16;
    const int row  = row0 + wave;

    int b   = row / (T_ * U1_);
    int rem = row - b * (T_ * U1_);
    int t   = rem / U1_;
    int u   = rem - t * U1_;

    const float* base = logits + (size_t)row * V1_;
    const int li = (u < U_) ? y[b * U_ + u] : -1;   // label column (>=1), -1 if none

    float4 v[8];
    float m = -INFINITY;
    #pragma unroll
    for (int k = 0; k < 8; ++k) {
        v[k] = *(const float4*)(base + k * 128 + lane * 4);
        m = fmaxf(m, fmaxf(fmaxf(v[k].x, v[k].y), fmaxf(v[k].z, v[k].w)));
    }
    // wave32 max reduction
    #pragma unroll
    for (int off = 16; off >= 1; off >>= 1)
        m = fmaxf(m, __shfl_xor(m, off, 32));

    float s = 0.f;
    #pragma unroll
    for (int k = 0; k < 8; ++k) {
        s += expf(v[k].x - m) + expf(v[k].y - m) + expf(v[k].z - m) + expf(v[k].w - m);
        const int basec = k * 128 + lane * 4;
        if (li >= basec && li < basec + 4) {
            float val = (li == basec)     ? v[k].x
                      : (li == basec + 1) ? v[k].y
                      : (li == basec + 2) ? v[k].z : v[k].w;
            sLab[wave] = val;               // exactly one lane writes
        }
    }
    sP[wave * 33 + lane] = s;
    if (lane == 0) { sMax[wave] = m; sBlank[wave] = v[0].x; }
    __syncthreads();

    // wave 0: rowsum[16] = P(16x32) x ones(32x1) via 8 WMMA f32 16x16x4 steps
    if (tid < 32) {
        v8f acc = {};
        v2f ones; ones[0] = 1.0f; ones[1] = 1.0f;   // B all-ones, any layout
        const int r  = lane & 15;                   // A row M
        const int kk = (lane >> 4) * 2;             // K pair: lanes0-15 -> K0,K1; 16-31 -> K2,K3
        #pragma unroll
        for (int c4 = 0; c4 < 8; ++c4) {
            v2f a;
            a[0] = sP[r * 33 + c4 * 4 + kk];
            a[1] = sP[r * 33 + c4 * 4 + kk + 1];
            acc = __builtin_amdgcn_wmma_f32_16x16x4_f32(
                false, a, false, ones, (short)0, acc, false, false);
        }
        #pragma unroll
        for (int vv = 0; vv < 8; ++vv) sD[vv * 32 + lane] = acc[vv];
    }
    __syncthreads();

    // D layout: VGPR v, lanes 0-15 => M=v ; lanes 16-31 => M=v+8 (all N equal)
    if (tid < 16) {
        const int mrow = tid;
        const float rs  = sD[(mrow & 7) * 32 + ((mrow >= 8) ? 16 : 0)];
        const float lse = sMax[mrow] + logf(rs);
        const int rr  = row0 + mrow;
        int b2   = rr / (T_ * U1_);
        int rem2 = rr - b2 * (T_ * U1_);
        int t2   = rem2 / U1_;
        int u2   = rem2 - t2 * U1_;
        lp_blank[rr] = sBlank[mrow] - lse;
        if (u2 < U_)
            lp_label[(size_t)b2 * T_ * U_ + (size_t)t2 * U_ + u2] = sLab[mrow] - lse;
    }
}

// ---------------------------------------------------------------------------
// Kernel 2: wavefront DP over anti-diagonals d = t+u. One block: 65 u-threads
// x 8 batches. Each diagonal depends only on the previous one, so 319 barrier
// steps total. Thread (0,0) writes the final -mean(log_like).
// ---------------------------------------------------------------------------
__global__ __launch_bounds__(520) void rnnt_dp_kernel(
    const float* __restrict__ lp_blank, const float* __restrict__ lp_label,
    const int* __restrict__ logit_lens, const int* __restrict__ y_lens,
    float* __restrict__ out)
{
    __shared__ float sh[B_][U1_];
    __shared__ float sll[B_];

    const int u = threadIdx.x;   // 0..64
    const int b = threadIdx.y;   // 0..7
    const float* blk = lp_blank + (size_t)b * T_ * U1_;
    const float* lab = lp_label + (size_t)b * T_ * U_;
    const int t_end = logit_lens[b] - 1;
    const int u_end = y_lens[b];

    // alpha[0,u] = cumsum of lp_label[b,0,k] for k < u
    float alpha = 0.f;
    for (int k = 0; k < u; ++k) alpha += lab[k];
    if (t_end == 0 && u == u_end) sll[b] = alpha + blk[u];
    sh[b][u] = alpha;
    __syncthreads();

    for (int d = 1; d <= (T_ - 1) + U_; ++d) {
        const float nb = (u > 0) ? sh[b][u - 1] : 0.f;   // neighbor from diag d-1
        __syncthreads();
        const int t = d - u;
        if (t >= 1 && t < T_) {
            const float fb = alpha + blk[(size_t)(t - 1) * U1_ + u];
            if (u == 0) alpha = fb;
            else        alpha = logaddexpf_(fb, nb + lab[(size_t)t * U_ + (u - 1)]);
            sh[b][u] = alpha;
            if (t == t_end && u == u_end)
                sll[b] = alpha + blk[(size_t)t * U1_ + u];
        }
        __syncthreads();
    }

    if (u == 0 && b == 0) {
        float s = 0.f;
        for (int i = 0; i < B_; ++i) s += sll[i];
        out[0] = -s / (float)B_;
    }
}

extern "C" void kernel_launch(void* const* d_in, const int* in_sizes, int n_in,
                              void* d_out, int out_size, void* d_ws, size_t ws_size,
                              hipStream_t stream) {
    const float* logits     = (const float*)d_in[0];
    const int*   logit_lens = (const int*)d_in[1];
    const int*   y          = (const int*)d_in[2];
    const int*   y_lens     = (const int*)d_in[3];
    float* out = (float*)d_out;

    float* lp_blank = (float*)d_ws;                 // [B,T,U+1]  133120 floats
    float* lp_label = lp_blank + ROWS_;             // [B,T,U]    131072 floats

    rnnt_logsoftmax_kernel<<<ROWS_ / 16, 512, 0, stream>>>(logits, y, lp_blank, lp_label);
    rnnt_dp_kernel<<<1, dim3(U1_, B_), 0, stream>>>(lp_blank, lp_label, logit_lens, y_lens, out);
}